// TreeOfThoughts_43628277793145
// MI455X (gfx1250) — compile-verified
//
#include <hip/hip_runtime.h>
#include <math.h>

// ---------------- problem constants ----------------
#define HD      512          // H
#define HD2     1024         // 2H
#define S_TOT   2048
#define LDA     516          // LDS row stride for H-wide arrays (bank-conflict pad)
#define LDH1    1028         // LDS row stride for 2H-wide array
#define NWAVES  8
#define TEMP_INV 1.25f       // 1/0.8
#define MIN_DIFF 0.1f
#define BEAM_SZ (16 * LDA)

typedef __attribute__((ext_vector_type(2)))  float v2f;
typedef __attribute__((ext_vector_type(8)))  float v8f;

// LDS map (all uses stay sm+offset expressions -> addrspace(3) inferred):
//   S_BEAM(0/1): beams, candidate-th scratch, h2 scratch (time-multiplexed)
//   S_PROJ(0/1): per-beam projections (live across each depth)
//   h1A/h1B    : evaluator hidden pair
#define S_BEAM(j) (sm + (j) * BEAM_SZ)
#define S_PROJ(k) (sm + (2 + (k)) * BEAM_SZ)

__device__ __forceinline__ float gelu_exact(float x) {
    return 0.5f * x * (1.0f + erff(x * 0.70710678118654752440f));
}

// ---- store one 16x16 C tile to LDS with bias (+optional exact gelu), row-masked ----
__device__ __forceinline__ void store_tile(float* dst, int ldd, int n0, v8f c,
                                           const float* __restrict__ bias,
                                           int act, int lane, int rowlo, int rowhi)
{
    const int col   = lane & 15;
    const int mbase = (lane & 16) ? 8 : 0;
    const float bv  = bias[n0 + col];
    #pragma unroll
    for (int r = 0; r < 8; r++) {
        int m = mbase + r;                 // C/D layout: VGPR r -> M = r (+8 hi half)
        if (m < rowlo || m > rowhi) continue;
        float v = c[r] + bv;
        if (act) v = gelu_exact(v);
        dst[m * ldd + n0 + col] = v;
    }
}

// ---- single-M: 16xN = (16xK LDS)@(KxN global), two N-tiles per wave per K-sweep ----
// A (f32 16x4): lane holds row lane%16; VGPR0/1 = K = (lane>=16?2:0)+{0,1}
// B (4x16): lane holds column, same K split.
__device__ __forceinline__ void mm_full(const float* __restrict__ A, int lda,
                                        const float* __restrict__ B, int ldb,
                                        int N, int K, float* dst, int ldd,
                                        const float* __restrict__ bias, int act,
                                        int rowlo, int rowhi, int wave, int lane)
{
    const int row = lane & 15;
    const int col = lane & 15;
    const int kb  = (lane & 16) ? 2 : 0;
    const float* arow = A + row * lda + kb;
    const int nt = N >> 4;
    for (int t = wave; t < nt; t += 2 * NWAVES) {
        const int n0a = t << 4;
        const int n0b = (t + NWAVES) << 4;
        const float* bA = B + (size_t)kb * ldb + n0a + col;
        const float* bB = B + (size_t)kb * ldb + n0b + col;
        v8f ca = {0.f,0.f,0.f,0.f,0.f,0.f,0.f,0.f};
        v8f cb = {0.f,0.f,0.f,0.f,0.f,0.f,0.f,0.f};
        #pragma unroll 8
        for (int k0 = 0; k0 < K; k0 += 4) {
            if ((k0 & 31) == 0 && (k0 + 32) < K) {
                __builtin_prefetch(&bA[(size_t)(k0 + 32) * ldb], 0, 1);
                __builtin_prefetch(&bB[(size_t)(k0 + 32) * ldb], 0, 1);
            }
            v2f a  = { arow[k0], arow[k0 + 1] };
            v2f b0 = { bA[(size_t)k0 * ldb], bA[(size_t)(k0 + 1) * ldb] };
            v2f b1 = { bB[(size_t)k0 * ldb], bB[(size_t)(k0 + 1) * ldb] };
            ca = __builtin_amdgcn_wmma_f32_16x16x4_f32(false, a, false, b0,
                                                       (short)0, ca, false, false);
            cb = __builtin_amdgcn_wmma_f32_16x16x4_f32(false, a, false, b1,
                                                       (short)0, cb, false, false);
        }
        store_tile(dst, ldd, n0a, ca, bias, act, lane, rowlo, rowhi);
        store_tile(dst, ldd, n0b, cb, bias, act, lane, rowlo, rowhi);
    }
}

// ---- dual-M: 32xN = two 16-row A slabs sharing the SAME weight stream ----
// 2M x 2N register tile: 4 accumulators; each B load feeds 2 wmma, each A load 2 wmma.
// Halves weight (global) traffic vs running the two slabs separately.
__device__ __forceinline__ void mm_dual(const float* __restrict__ A0,
                                        const float* __restrict__ A1, int lda,
                                        const float* __restrict__ B, int ldb,
                                        int N, int K,
                                        float* d0, float* d1, int ldd,
                                        const float* __restrict__ bias, int act,
                                        int wave, int lane)
{
    const int row = lane & 15;
    const int col = lane & 15;
    const int kb  = (lane & 16) ? 2 : 0;
    const float* a0row = A0 + row * lda + kb;
    const float* a1row = A1 + row * lda + kb;
    const int nt = N >> 4;
    for (int t = wave; t < nt; t += 2 * NWAVES) {
        const int n0a = t << 4;
        const int n0b = (t + NWAVES) << 4;
        const float* bA = B + (size_t)kb * ldb + n0a + col;
        const float* bB = B + (size_t)kb * ldb + n0b + col;
        v8f c00 = {0.f,0.f,0.f,0.f,0.f,0.f,0.f,0.f};
        v8f c01 = {0.f,0.f,0.f,0.f,0.f,0.f,0.f,0.f};
        v8f c10 = {0.f,0.f,0.f,0.f,0.f,0.f,0.f,0.f};
        v8f c11 = {0.f,0.f,0.f,0.f,0.f,0.f,0.f,0.f};
        #pragma unroll 4
        for (int k0 = 0; k0 < K; k0 += 4) {
            if ((k0 & 31) == 0 && (k0 + 32) < K) {
                __builtin_prefetch(&bA[(size_t)(k0 + 32) * ldb], 0, 1);
                __builtin_prefetch(&bB[(size_t)(k0 + 32) * ldb], 0, 1);
            }
            v2f a0 = { a0row[k0], a0row[k0 + 1] };
            v2f a1 = { a1row[k0], a1row[k0 + 1] };
            v2f b0 = { bA[(size_t)k0 * ldb], bA[(size_t)(k0 + 1) * ldb] };
            v2f b1 = { bB[(size_t)k0 * ldb], bB[(size_t)(k0 + 1) * ldb] };
            c00 = __builtin_amdgcn_wmma_f32_16x16x4_f32(false, a0, false, b0,
                                                        (short)0, c00, false, false);
            c10 = __builtin_amdgcn_wmma_f32_16x16x4_f32(false, a1, false, b0,
                                                        (short)0, c10, false, false);
            c01 = __builtin_amdgcn_wmma_f32_16x16x4_f32(false, a0, false, b1,
                                                        (short)0, c01, false, false);
            c11 = __builtin_amdgcn_wmma_f32_16x16x4_f32(false, a1, false, b1,
                                                        (short)0, c11, false, false);
        }
        store_tile(d0, ldd, n0a, c00, bias, act, lane, 0, 15);
        store_tile(d0, ldd, n0b, c01, bias, act, lane, 0, 15);
        store_tile(d1, ldd, n0a, c10, bias, act, lane, 0, 15);
        store_tile(d1, ldd, n0b, c11, bias, act, lane, 0, 15);
    }
}

// ---- evaluator on a candidate PAIR (32 rows share one weight stream) ----
// thA/thB overwritten by h2; sSig[0..15] = cand A rows, sSig[16..31] = cand B rows.
__device__ __forceinline__ void evaluate_pair(float* thA, float* thB,
    float* h1A, float* h1B,
    const float* We1, const float* be1, const float* We2, const float* be2,
    const float* We3, const float* be3,
    float* sRed, float* sSig, int tid, int wave, int lane)
{
    mm_dual(thA, thB, LDA, We1, HD2, HD2, HD, h1A, h1B, LDH1, be1, 1, wave, lane);
    __syncthreads();
    mm_dual(h1A, h1B, LDH1, We2, HD, HD, HD2, thA, thB, LDA, be2, 1, wave, lane);
    __syncthreads();
    if (tid < 32) sRed[tid] = 0.f;
    __syncthreads();
    {   // 512 -> 1 projection: 32 rows x 8 chunks of 64
        int row = tid >> 3, ch = tid & 7;
        const float* src = (row < 16) ? (thA + row * LDA) : (thB + (row - 16) * LDA);
        const float* h2r = src + ch * 64;
        const float* w3  = We3 + ch * 64;
        float part = 0.f;
        #pragma unroll
        for (int i = 0; i < 64; i++) part += h2r[i] * w3[i];
        atomicAdd(&sRed[row], part);            // ds_add_f32
    }
    __syncthreads();
    if (tid < 32) {
        float l = sRed[tid] + be3[0];
        sSig[tid] = 1.0f / (1.0f + expf(-l));
    }
    __syncthreads();
}

// ---- reference keep/top-2 selection for one position (serial, tiny) ----
__device__ void select2(const float* sc, int K, int* selOut)
{
    const int M = K * 4;
    bool keep[8];
    for (int k = 0; k < K; k++) {
        int cnt = 0;
        for (int i = 0; i < 4; i++) {
            int idx = k * 4 + i;
            float mind = INFINITY;
            for (int j = 0; j < i; j++) {
                int jj = k * 4 + j;
                if (keep[jj]) mind = fminf(mind, fabsf(sc[idx] - sc[jj]));
            }
            bool ki = ((cnt == 0) || (mind >= MIN_DIFF)) && (cnt < 2);
            keep[idx] = ki;
            cnt += ki ? 1 : 0;
        }
    }
    int b0 = -1; float v0 = -INFINITY;
    for (int i = 0; i < M; i++) {
        float v = keep[i] ? sc[i] : -INFINITY;
        if (v > v0) { v0 = v; b0 = i; }           // ties -> lowest index (matches top_k)
    }
    int b1 = -1; float v1 = -INFINITY;
    for (int i = 0; i < M; i++) {
        if (i == b0) continue;
        float v = keep[i] ? sc[i] : -INFINITY;
        if (v > v1) { v1 = v; b1 = i; }
    }
    selOut[0] = b0;
    selOut[1] = (v1 > -INFINITY) ? b1 : -1;       // -1 == invalid slot1
}

// -------------------------- fused kernel: 2 positions / WG --------------------------
__global__ __launch_bounds__(256, 1)
void TreeOfThoughts_kernel(const float* __restrict__ hs,
                           const float* __restrict__ We1, const float* __restrict__ be1,
                           const float* __restrict__ We2, const float* __restrict__ be2,
                           const float* __restrict__ We3, const float* __restrict__ be3,
                           const float* __restrict__ Wp,  const float* __restrict__ bp,
                           const float* __restrict__ Wb,  const float* __restrict__ bb,
                           float* __restrict__ out)
{
    extern __shared__ float sm[];
    float* h1A     = sm + 4 * BEAM_SZ;
    float* h1B     = h1A + 16 * LDH1;
    float* sRed    = h1B + 16 * LDH1;         // 32
    float* sSig    = sRed + 32;               // 32
    float* sScores = sSig + 32;               // 2 positions x 8 candidates
    int*   sSel    = (int*)(sScores + 16);    // 2 positions x 2 slots

    const int tid  = threadIdx.x;
    const int lane = tid & 31;
    const int wave = tid >> 5;
    const int s0   = blockIdx.x * 2;          // rows 0..7 -> pos s0, rows 8..15 -> s0+1

    // beam0 = x0 (initial ev(x0) score is dead in the reference -> skipped)
    for (int i = tid; i < 16 * HD; i += 256) {
        int m = i >> 9, h = i & (HD - 1);
        int p = m >> 3, b = m & 7;
        sm[m * LDA + h] = hs[((size_t)(b * S_TOT + s0 + p)) * HD + h];
    }
    __syncthreads();

    int K = 1;
    for (int depth = 0; depth < 3; depth++) {
        if (K == 1) {
            // proj0 = beam0 @ Wp + bp
            mm_full(S_BEAM(0), LDA, Wp, HD, HD, HD, S_PROJ(0), LDA, bp, 0, 0, 15, wave, lane);
            __syncthreads();
            // 4 candidates (k=0, n=0..3): generate th pairs, evaluate pairwise
            for (int pr = 0; pr < 2; pr++) {
                int na = pr * 2, nb = pr * 2 + 1;
                mm_full(S_PROJ(0), LDA, Wb + (size_t)na * HD * HD, HD, HD, HD,
                        S_BEAM(0), LDA, bb + na * HD, 1, 0, 15, wave, lane);
                mm_full(S_PROJ(0), LDA, Wb + (size_t)nb * HD * HD, HD, HD, HD,
                        S_BEAM(1), LDA, bb + nb * HD, 1, 0, 15, wave, lane);
                __syncthreads();
                evaluate_pair(S_BEAM(0), S_BEAM(1), h1A, h1B,
                              We1, be1, We2, be2, We3, be3, sRed, sSig, tid, wave, lane);
                if (tid < 2) {
                    float aA = 0.f, aB = 0.f;
                    #pragma unroll
                    for (int b = 0; b < 8; b++) { aA += sSig[tid*8 + b]; aB += sSig[16 + tid*8 + b]; }
                    sScores[tid * 8 + na] = (aA * 0.125f) * TEMP_INV;
                    sScores[tid * 8 + nb] = (aB * 0.125f) * TEMP_INV;
                }
                __syncthreads();
            }
        } else {
            // both beams project against the same Wp stream (M=32)
            mm_dual(S_BEAM(0), S_BEAM(1), LDA, Wp, HD, HD, HD,
                    S_PROJ(0), S_PROJ(1), LDA, bp, 0, wave, lane);
            __syncthreads();
            // candidates (k=0,n) and (k=1,n) share Wb[n] -> M=32 everywhere
            for (int n = 0; n < 4; n++) {
                mm_dual(S_PROJ(0), S_PROJ(1), LDA, Wb + (size_t)n * HD * HD, HD, HD, HD,
                        S_BEAM(0), S_BEAM(1), LDA, bb + n * HD, 1, wave, lane);
                __syncthreads();
                evaluate_pair(S_BEAM(0), S_BEAM(1), h1A, h1B,
                              We1, be1, We2, be2, We3, be3, sRed, sSig, tid, wave, lane);
                if (tid < 2) {
                    float aA = 0.f, aB = 0.f;
                    #pragma unroll
                    for (int b = 0; b < 8; b++) { aA += sSig[tid*8 + b]; aB += sSig[16 + tid*8 + b]; }
                    sScores[tid * 8 + n]     = (aA * 0.125f) * TEMP_INV;   // cand (k=0,n)
                    sScores[tid * 8 + 4 + n] = (aB * 0.125f) * TEMP_INV;   // cand (k=1,n)
                }
                __syncthreads();
            }
        }

        // per-position keep/top-2
        if (tid < 2) select2(sScores + tid * 8, K, sSel + tid * 2);
        __syncthreads();

        // recompute winning th states into the beam buffers (row-masked per position)
        for (int j = 0; j < 2; j++) {
            int c0 = sSel[0 * 2 + j];
            int c1 = sSel[1 * 2 + j];
            if (c0 >= 0 && c0 == c1) {
                int k = c0 >> 2, n = c0 & 3;
                mm_full(S_PROJ(k), LDA, Wb + (size_t)n * HD * HD, HD, HD, HD,
                        S_BEAM(j), LDA, bb + n * HD, 1, 0, 15, wave, lane);
            } else {
                if (c0 >= 0) {
                    int k = c0 >> 2, n = c0 & 3;
                    mm_full(S_PROJ(k), LDA, Wb + (size_t)n * HD * HD, HD, HD, HD,
                            S_BEAM(j), LDA, bb + n * HD, 1, 0, 7, wave, lane);
                }
                if (c1 >= 0) {
                    int k = c1 >> 2, n = c1 & 3;
                    mm_full(S_PROJ(k), LDA, Wb + (size_t)n * HD * HD, HD, HD, HD,
                            S_BEAM(j), LDA, bb + n * HD, 1, 8, 15, wave, lane);
                }
            }
            __syncthreads();
        }
        // invalid slot1 -> duplicate new slot0 (reference: top_st[:, :1] fallback)
        for (int p = 0; p < 2; p++) {
            if (sSel[p * 2 + 1] < 0) {
                for (int i = tid; i < 8 * HD; i += 256) {
                    int m = p * 8 + (i >> 9), h = i & (HD - 1);
                    sm[BEAM_SZ + m * LDA + h] = sm[m * LDA + h];
                }
            }
        }
        __syncthreads();
        K = 2;
    }

    // out = best beam + residual
    for (int i = tid; i < 16 * HD; i += 256) {
        int m = i >> 9, h = i & (HD - 1);
        int p = m >> 3, b = m & 7;
        size_t g = ((size_t)(b * S_TOT + s0 + p)) * HD + h;
        out[g] = sm[m * LDA + h] + hs[g];
    }
}

// ---------------------------------- launch ----------------------------------
extern "C" void kernel_launch(void* const* d_in, const int* in_sizes, int n_in,
                              void* d_out, int out_size, void* d_ws, size_t ws_size,
                              hipStream_t stream)
{
    (void)in_sizes; (void)n_in; (void)out_size; (void)d_ws; (void)ws_size;
    const float* hs  = (const float*)d_in[0];
    const float* We1 = (const float*)d_in[1];
    const float* be1 = (const float*)d_in[2];
    const float* We2 = (const float*)d_in[3];
    const float* be2 = (const float*)d_in[4];
    const float* We3 = (const float*)d_in[5];
    const float* be3 = (const float*)d_in[6];
    const float* Wp  = (const float*)d_in[7];
    const float* bp  = (const float*)d_in[8];
    const float* Wb  = (const float*)d_in[9];
    const float* bb  = (const float*)d_in[10];
    float* out = (float*)d_out;

    const size_t smem = (size_t)(4 * BEAM_SZ + 2 * 16 * LDH1 + 32 + 32 + 16 + 4) * sizeof(float);
    hipFuncSetAttribute((const void*)TreeOfThoughts_kernel,
                        hipFuncAttributeMaxDynamicSharedMemorySize, (int)smem);

    TreeOfThoughts_kernel<<<dim3(S_TOT / 2), dim3(256), smem, stream>>>(
        hs, We1, be1, We2, be2, We3, be3, Wp, bp, Wb, bb, out);
}